// SAMGIPool_8684423872566
// MI455X (gfx1250) — compile-verified
//
#include <hip/hip_runtime.h>
#include <hip/hip_bf16.h>

// ---------------- problem constants ----------------
constexpr int N    = 50000;
constexpr int H    = 128;
constexpr int E    = 1600000;
constexpr int K    = 25000;            // ceil(0.5 * N)
constexpr int MT   = E + K;            // edges + self loops
constexpr int NP   = 65536;            // pow2 pad for bitonic top-k
constexpr float LAMB      = 1.0f;
constexpr float NEG_SLOPE = 0.2f;

typedef __attribute__((ext_vector_type(2))) float v2f;
typedef __attribute__((ext_vector_type(8))) float v8f;
typedef unsigned long long ull;

__device__ __forceinline__ float sigm(float x) { return 1.0f / (1.0f + expf(-x)); }
__device__ __forceinline__ float lrelu(float x) { return x > 0.0f ? x : x * NEG_SLOPE; }

// monotone float -> uint map (larger float => larger uint)
__device__ __forceinline__ unsigned fkey(float f) {
  unsigned u = __float_as_uint(f);
  return (u & 0x80000000u) ? ~u : (u | 0x80000000u);
}

// ---------------- stage 1: fused view-attention node scores ----------------
__global__ void k_scores(const float* __restrict__ sc1, const float* __restrict__ sc2,
                         const float* __restrict__ va, const float* __restrict__ vb,
                         float* __restrict__ scores) {
  int i = blockIdx.x * blockDim.x + threadIdx.x;
  if (i >= N) return;
  float a = sc1[i], b = sc2[i];
  // scores_cat @ view_att + bias, row-major view_att[2][2]
  float t0 = sigm(a * va[0] + b * va[2] + vb[0]);
  float t1 = sigm(a * va[1] + b * va[3] + vb[1]);
  float m  = fmaxf(t0, t1);
  float e0 = expf(t0 - m), e1 = expf(t1 - m);
  float inv = 1.0f / (e0 + e1);
  scores[i] = sigm(a * (e0 * inv) + b * (e1 * inv));
}

// ---------------- stage 2: init keys / node_map / segment arrays ----------------
__global__ void k_init(const float* __restrict__ scores, ull* __restrict__ keys,
                       int* __restrict__ node_map, int* __restrict__ seg_count,
                       int* __restrict__ seg_cursor) {
  int i = blockIdx.x * blockDim.x + threadIdx.x;
  if (i < NP) {
    ull key;
    if (i < N) {
      unsigned u = fkey(scores[i]);
      key = ((ull)(~u) << 32) | (unsigned)i;   // ascending key == descending score, idx tie-break
    } else {
      key = ~0ull;                              // pad sinks to the end
    }
    keys[i] = key;
  }
  if (i < N) node_map[i] = -1;
  if (i < K) { seg_count[i] = 0; seg_cursor[i] = 0; }
}

// global bitonic pass over NP keys (ascending)
__global__ void k_bitonic(ull* __restrict__ keys, int k, int j) {
  int i = blockIdx.x * blockDim.x + threadIdx.x;
  int ixj = i ^ j;
  if (ixj > i && ixj < NP && i < NP) {
    ull a = keys[i], b = keys[ixj];
    bool asc = ((i & k) == 0);
    if (asc ? (a > b) : (a < b)) { keys[i] = b; keys[ixj] = a; }
  }
}

// ---------------- stage 3: gather xp = x[perm]*score, scatter node_map ----------------
__global__ void k_gather(const ull* __restrict__ keys, const float* __restrict__ x,
                         const float* __restrict__ scores, float* __restrict__ xp,
                         float* __restrict__ out_perm, int* __restrict__ node_map) {
  int i = blockIdx.x * blockDim.x + threadIdx.x;
  if (i >= K * H) return;
  int k = i >> 7;            // H == 128
  int h = i & (H - 1);
  int idx = (int)(keys[k] & 0xffffffffu);
  float ts = scores[idx];
  xp[i] = x[idx * H + h] * ts;
  if (h == 0) {
    out_perm[k] = (float)idx;
    node_map[idx] = k;
  }
}

// ---------------- stage 4: dual matvec s1/s2 via V_WMMA_F32_16X16X4_F32 ----------------
// One wave handles a 16-row tile of xp; B holds the two struct_att columns
// (N=0 -> struct_att[0:H], N=1 -> struct_att[H:2H]); accumulate K=128 in chunks of 4.
__global__ void k_gemv_wmma(const float* __restrict__ xp, const float* __restrict__ sa,
                            float* __restrict__ s1, float* __restrict__ s2) {
  int wave = (blockIdx.x * blockDim.x + threadIdx.x) >> 5;
  int lane = threadIdx.x & 31;
  int m0 = wave * 16;
  if (m0 >= K) return;                          // wave-uniform exit (EXEC stays full)
  int mrow  = m0 + (lane & 15);
  int kbase = (lane >> 4) << 1;                 // K = {0,1} lanes 0-15, {2,3} lanes 16-31
  int n     = lane & 15;                        // B column / D column index
  bool mv   = (mrow < K);
  const float* arow = xp + (size_t)(mv ? mrow : (K - 1)) * H;
  v8f c = {};
#pragma unroll 4
  for (int h0 = 0; h0 < H; h0 += 4) {
    v2f a, b;
    a.x = mv ? arow[h0 + kbase]     : 0.0f;
    a.y = mv ? arow[h0 + kbase + 1] : 0.0f;
    b.x = (n < 2) ? sa[n * H + h0 + kbase]     : 0.0f;
    b.y = (n < 2) ? sa[n * H + h0 + kbase + 1] : 0.0f;
    c = __builtin_amdgcn_wmma_f32_16x16x4_f32(false, a, false, b, (short)0, c, false, false);
  }
  // D layout: VGPR j -> (M=j, N=lane) lanes 0-15 ; (M=j+8, N=lane-16) lanes 16-31
  if (n < 2) {
    float* dst = (n == 0) ? s1 : s2;
    int mbase = m0 + ((lane >> 4) << 3);
#pragma unroll
    for (int j = 0; j < 8; ++j)
      if (mbase + j < K) dst[mbase + j] = c[j];
  }
}

// ---------------- stage 5: edge remap + weights + edge-index output ----------------
__global__ void k_edges(const int* __restrict__ erow, const int* __restrict__ ecol,
                        const float* __restrict__ eattr, const int* __restrict__ node_map,
                        const float* __restrict__ s1, const float* __restrict__ s2,
                        int* __restrict__ rows, int* __restrict__ cols,
                        float* __restrict__ w, float* __restrict__ out_ei) {
  int e = blockIdx.x * blockDim.x + threadIdx.x;
  if (e >= MT) return;
  int rr, cc; float wv;
  if (e < E) {
    int nr = node_map[erow[e]];
    int nc = node_map[ecol[e]];
    bool valid = (nr >= 0) && (nc >= 0);
    rr = valid ? nr : K;
    cc = valid ? nc : K;
    float at  = valid ? eattr[e] : 0.0f;
    float raw = s1[rr < K ? rr : K - 1] + s2[cc < K ? cc : K - 1];
    wv = valid ? (lrelu(raw) + at * LAMB) : 0.0f;
  } else {
    int i = e - E;                       // self loop, attr 0, mask true
    rr = i; cc = i;
    wv = lrelu(s1[i] + s2[i]);
  }
  rows[e] = rr; cols[e] = cc; w[e] = wv;
  out_ei[e]      = (float)rr;            // new_edge_index row 0
  out_ei[MT + e] = (float)cc;            // new_edge_index row 1
}

// ---------------- stage 6a: per-segment histogram (skip dummy segment K) ----------------
__global__ void k_hist(const int* __restrict__ rows, int* __restrict__ seg_count) {
  int e = blockIdx.x * blockDim.x + threadIdx.x;
  if (e >= MT) return;
  int r = rows[e];
  if (r < K) atomicAdd(&seg_count[r], 1);
}

// ---------------- stage 6b: single-block exclusive scan over K counts ----------------
__global__ void k_scan(const int* __restrict__ counts, int* __restrict__ starts) {
  __shared__ int temp[1024];
  __shared__ int carry;
  int tid = threadIdx.x;
  if (tid == 0) carry = 0;
  __syncthreads();
  for (int base = 0; base < K; base += 1024) {
    int c0 = carry;
    int v = (base + tid < K) ? counts[base + tid] : 0;
    temp[tid] = v;
    __syncthreads();
    for (int off = 1; off < 1024; off <<= 1) {
      int t = (tid >= off) ? temp[tid - off] : 0;
      __syncthreads();
      temp[tid] += t;
      __syncthreads();
    }
    if (base + tid < K) starts[base + tid] = c0 + temp[tid] - v;   // exclusive
    __syncthreads();
    if (tid == 0) carry = c0 + temp[1023];
    __syncthreads();
  }
}

// ---------------- stage 6c: scatter weights into per-segment buckets ----------------
__global__ void k_scatter(const int* __restrict__ rows, const float* __restrict__ w,
                          const int* __restrict__ seg_start, int* __restrict__ seg_cursor,
                          float* __restrict__ segvals) {
  int e = blockIdx.x * blockDim.x + threadIdx.x;
  if (e >= MT) return;
  int r = rows[e];
  if (r < K) {
    int pos = seg_start[r] + atomicAdd(&seg_cursor[r], 1);
    segvals[pos] = w[e];
  }
}

// ---------------- stage 7: per-segment descending sort + sparsemax tau ----------------
__global__ void k_tau(const float* __restrict__ segvals, const int* __restrict__ seg_start,
                      const int* __restrict__ seg_count, float* __restrict__ tau) {
  __shared__ float z[1024];
  int r = blockIdx.x;                      // r in [0, K)
  int cnt = seg_count[r];
  int start = seg_start[r];
  if (cnt > 1024) cnt = 1024;              // Poisson(~17) segments; cap is enormous margin
  int P = 2; while (P < cnt) P <<= 1;
  for (int i = threadIdx.x; i < P; i += blockDim.x)
    z[i] = (i < cnt) ? segvals[start + i] : -INFINITY;
  __syncthreads();
  for (int kk = 2; kk <= P; kk <<= 1) {
    for (int j = kk >> 1; j > 0; j >>= 1) {
      for (int i = threadIdx.x; i < P; i += blockDim.x) {
        int ixj = i ^ j;
        if (ixj > i) {
          float a = z[i], b = z[ixj];
          bool desc = ((i & kk) == 0);
          if (desc ? (a < b) : (a > b)) { z[i] = b; z[ixj] = a; }
        }
      }
      __syncthreads();
    }
  }
  if (threadIdx.x == 0) {
    if (cnt == 0) { tau[r] = 0.0f; return; }
    float cum = 0.0f, cumk = 0.0f; int ksel = 1;
    for (int i = 0; i < cnt; ++i) {
      cum += z[i];
      if (1.0f + (float)(i + 1) * z[i] > cum) { ksel = i + 1; cumk = cum; }
    }
    tau[r] = (cumk - 1.0f) / (float)ksel;
  }
}

// ---------------- stage 8: final masked sparsemax projection ----------------
__global__ void k_final(const int* __restrict__ rows, const float* __restrict__ w,
                        const float* __restrict__ tau, float* __restrict__ out_attr) {
  int e = blockIdx.x * blockDim.x + threadIdx.x;
  if (e >= MT) return;
  int r = rows[e];
  float p = 0.0f;
  if (r < K) {                             // mask == (segment != dummy)
    float v = w[e] - tau[r];
    p = v > 0.0f ? v : 0.0f;
  }
  out_attr[e] = p;
}

// ---------------- host orchestration ----------------
extern "C" void kernel_launch(void* const* d_in, const int* in_sizes, int n_in,
                              void* d_out, int out_size, void* d_ws, size_t ws_size,
                              hipStream_t stream) {
  const float* x     = (const float*)d_in[0];
  const int*   ei    = (const int*)d_in[1];
  const float* eattr = (const float*)d_in[2];
  const float* sc1   = (const float*)d_in[3];
  const float* sc2   = (const float*)d_in[4];
  const float* sa    = (const float*)d_in[5];   // struct_att [1, 2H]
  const float* va    = (const float*)d_in[6];   // view_att [2,2]
  const float* vb    = (const float*)d_in[7];   // view_bias [2]
  const int* erow = ei;
  const int* ecol = ei + E;

  // output layout (floats): xp | new_edge_index(2*MT) | new_edge_attr(MT) | perm(K)
  float* out_xp   = (float*)d_out;
  float* out_ei   = out_xp + (size_t)K * H;
  float* out_attr = out_ei + 2 * (size_t)MT;
  float* out_perm = out_attr + (size_t)MT;

  // workspace layout
  char* ws = (char*)d_ws;
  size_t off = 0;
  auto alloc = [&](size_t bytes) { char* p = ws + off; off = (off + bytes + 255) & ~(size_t)255; return p; };
  ull*   keys       = (ull*)  alloc((size_t)NP * 8);
  float* scores     = (float*)alloc((size_t)N * 4);
  int*   node_map   = (int*)  alloc((size_t)N * 4);
  float* s1         = (float*)alloc((size_t)K * 4);
  float* s2         = (float*)alloc((size_t)K * 4);
  int*   rows       = (int*)  alloc((size_t)MT * 4);
  int*   cols       = (int*)  alloc((size_t)MT * 4);
  float* w          = (float*)alloc((size_t)MT * 4);
  int*   seg_count  = (int*)  alloc((size_t)K * 4);
  int*   seg_start  = (int*)  alloc((size_t)K * 4);
  int*   seg_cursor = (int*)  alloc((size_t)K * 4);
  float* segvals    = (float*)alloc((size_t)MT * 4);
  float* tau        = (float*)alloc((size_t)K * 4);
  (void)ws_size; (void)in_sizes; (void)n_in; (void)out_size;

  const int B = 256;
  // 1. node scores
  k_scores<<<(N + B - 1) / B, B, 0, stream>>>(sc1, sc2, va, vb, scores);
  // 2. init + exact top-k via 64-bit bitonic sort (descending score, ascending idx tie-break)
  k_init<<<NP / B, B, 0, stream>>>(scores, keys, node_map, seg_count, seg_cursor);
  for (int k = 2; k <= NP; k <<= 1)
    for (int j = k >> 1; j > 0; j >>= 1)
      k_bitonic<<<NP / B, B, 0, stream>>>(keys, k, j);
  // 3. gather pooled features, scatter node_map, emit perm
  k_gather<<<(K * H + B - 1) / B, B, 0, stream>>>(keys, x, scores, out_xp, out_perm, node_map);
  // 4. s1/s2 dual matvec on WMMA (f32, exact)
  {
    int waves  = (K + 15) / 16;
    int blocks = (waves + 7) / 8;          // 8 waves / 256-thread block
    k_gemv_wmma<<<blocks, 256, 0, stream>>>(out_xp, sa, s1, s2);
  }
  // 5. edge remap + structural weights + edge-index output
  k_edges<<<(MT + B - 1) / B, B, 0, stream>>>(erow, ecol, eattr, node_map, s1, s2,
                                              rows, cols, w, out_ei);
  // 6. segment grouping: histogram -> scan -> scatter
  k_hist<<<(MT + B - 1) / B, B, 0, stream>>>(rows, seg_count);
  k_scan<<<1, 1024, 0, stream>>>(seg_count, seg_start);
  k_scatter<<<(MT + B - 1) / B, B, 0, stream>>>(rows, w, seg_start, seg_cursor, segvals);
  // 7. per-segment sparsemax threshold
  k_tau<<<K, 128, 0, stream>>>(segvals, seg_start, seg_count, tau);
  // 8. final masked projection
  k_final<<<(MT + B - 1) / B, B, 0, stream>>>(rows, w, tau, out_attr);
}